// SegmentIDEncoder_46737834115412
// MI455X (gfx1250) — compile-verified
//
#include <hip/hip_runtime.h>
#include <stdint.h>

// ---------------------------------------------------------------------------
// SegmentIDEncoder: gather + sorted-segment mean pool + L2 normalize.
// Memory-bound (~90 MB critical traffic -> ~4us roofline @ 23.3 TB/s HBM).
// CDNA5 paths used: global_load_async_to_lds_b128 (ASYNCcnt) staging of the
// per-block seg_id slice, s_wait_asynccnt, b128 vector loads/stores.
// ---------------------------------------------------------------------------

#ifndef __has_builtin
#define __has_builtin(x) 0
#endif

#if __has_builtin(__builtin_amdgcn_global_load_async_to_lds_b128)
#define SEGENC_ASYNC_LDS 1
#else
#define SEGENC_ASYNC_LDS 0
#endif

#if __has_builtin(__builtin_amdgcn_s_wait_asynccnt)
#define SEGENC_WAIT_ASYNC() __builtin_amdgcn_s_wait_asynccnt(0)
#else
#define SEGENC_WAIT_ASYNC() asm volatile("s_wait_asynccnt 0" ::: "memory")
#endif

typedef int segenc_v4i __attribute__((vector_size(16)));

#define VOX_PER_BLK 256
#define STAGE_CAP   12288   // ints of LDS staging (48 KB); expected need ~2K

__device__ __forceinline__ int lower_bound_i32(const int* __restrict__ a,
                                               int n, int key) {
  int lo = 0, hi = n;
  while (lo < hi) {
    int mid = (lo + hi) >> 1;
    if (a[mid] < key) lo = mid + 1; else hi = mid;
  }
  return lo;
}

__global__ __launch_bounds__(VOX_PER_BLK)
void segenc_pool_norm_kernel(const float* __restrict__ table,
                             const int*   __restrict__ seg_ids,
                             const int*   __restrict__ voxel_ids,
                             int T, int nvox,
                             float* __restrict__ out) {
  __shared__ int starts[VOX_PER_BLK + 1];
  __shared__ int sbuf[STAGE_CAP];

  const int tid   = threadIdx.x;
  const int vbase = blockIdx.x * VOX_PER_BLK;
  const int v     = vbase + tid;

  // ---- token-range starts for each voxel in this block (sorted voxel_ids) --
  {
    int key = v < nvox ? v : nvox;   // clamp: ids are < nvox, key==nvox -> T
    starts[tid] = lower_bound_i32(voxel_ids, T, key);
  }
  if (tid == 0) {
    int key = vbase + VOX_PER_BLK;
    if (key > nvox) key = nvox;
    starts[VOX_PER_BLK] = lower_bound_i32(voxel_ids, T, key);
  }
  __syncthreads();

  const int blkLo  = starts[0];
  const int blkHi  = starts[VOX_PER_BLK];
  const int g0     = blkLo & ~3;                  // 16B-align global start
  const int padded = ((blkHi - g0) + 3) & ~3;     // ints, multiple of 4
  const bool use_lds = (padded > 0) && (padded <= STAGE_CAP);

  // ---- stage this block's seg_id slice into LDS via async copies ----------
  if (use_lds) {
    const int nchunks = padded >> 2;              // 16-byte transfers
#if SEGENC_ASYNC_LDS
    for (int i = tid; i < nchunks; i += VOX_PER_BLK) {
      __builtin_amdgcn_global_load_async_to_lds_b128(
          (segenc_v4i*)(seg_ids + g0 + 4 * i),
          (segenc_v4i*)(&sbuf[4 * i]),
          0, 0);
    }
    SEGENC_WAIT_ASYNC();
#else
    const int4* gs = reinterpret_cast<const int4*>(seg_ids + g0);
    int4*       ls = reinterpret_cast<int4*>(sbuf);
    for (int i = tid; i < nchunks; i += VOX_PER_BLK) ls[i] = gs[i];
#endif
  }
  __syncthreads();

  if (v >= nvox) return;

  const int myLo = starts[tid];
  const int myHi = starts[tid + 1];

  // ---- accumulate this voxel's token embeddings (table rows are 48B = 3x16B)
  float acc[12];
#pragma unroll
  for (int d = 0; d < 12; ++d) acc[d] = 0.0f;

  for (int t = myLo; t < myHi; ++t) {
    const int seg = use_lds ? sbuf[t - g0] : seg_ids[t];
    const float4* row = reinterpret_cast<const float4*>(table + (size_t)seg * 12);
    float4 r0 = row[0], r1 = row[1], r2 = row[2];
    acc[0] += r0.x; acc[1] += r0.y; acc[2]  += r0.z; acc[3]  += r0.w;
    acc[4] += r1.x; acc[5] += r1.y; acc[6]  += r1.z; acc[7]  += r1.w;
    acc[8] += r2.x; acc[9] += r2.y; acc[10] += r2.z; acc[11] += r2.w;
  }

  // ---- mean + L2 normalize (divide by max(norm, 1e-12), like F.normalize) --
  const int   cnt = myHi - myLo;
  const float inv = 1.0f / (float)(cnt > 1 ? cnt : 1);
  float m[12], ss = 0.0f;
#pragma unroll
  for (int d = 0; d < 12; ++d) { m[d] = acc[d] * inv; ss += m[d] * m[d]; }
  const float scale = 1.0f / fmaxf(sqrtf(ss), 1e-12f);

  float4 o0 = make_float4(m[0] * scale, m[1] * scale, m[2]  * scale, m[3]  * scale);
  float4 o1 = make_float4(m[4] * scale, m[5] * scale, m[6]  * scale, m[7]  * scale);
  float4 o2 = make_float4(m[8] * scale, m[9] * scale, m[10] * scale, m[11] * scale);
  float4* orow = reinterpret_cast<float4*>(out + (size_t)v * 12);
  orow[0] = o0; orow[1] = o1; orow[2] = o2;
}

extern "C" void kernel_launch(void* const* d_in, const int* in_sizes, int n_in,
                              void* d_out, int out_size, void* d_ws, size_t ws_size,
                              hipStream_t stream) {
  const float* table     = (const float*)d_in[0];
  const int*   seg_ids   = (const int*)d_in[1];
  const int*   voxel_ids = (const int*)d_in[2];
  // d_in[3] (n_voxels) lives on device; derive it from the output size instead.
  const int T    = in_sizes[1];
  const int nvox = out_size / 12;

  dim3 grid((nvox + VOX_PER_BLK - 1) / VOX_PER_BLK), block(VOX_PER_BLK);
  segenc_pool_norm_kernel<<<grid, block, 0, stream>>>(
      table, seg_ids, voxel_ids, T, nvox, (float*)d_out);
}